// AttnDecoderRNN_14714557956454
// MI455X (gfx1250) — compile-verified
//
#include <hip/hip_runtime.h>
#include <math.h>

#define HID   1024
#define VOCAB 50257
#define SEQ   512

typedef __attribute__((ext_vector_type(2))) float        v2f;
typedef __attribute__((ext_vector_type(8))) float        v8f;
typedef __attribute__((ext_vector_type(4))) unsigned int u32x4;
typedef __attribute__((ext_vector_type(8))) int          i32x8;
typedef __attribute__((ext_vector_type(4))) int          i32x4;

#if __has_builtin(__builtin_amdgcn_tensor_load_to_lds)
#define HAS_TDM 1
#endif

// ---------------------------------------------------------------------------
// helpers
// ---------------------------------------------------------------------------
typedef __attribute__((address_space(3))) char lds_char_t;

__device__ __forceinline__ unsigned lds_byte_off(const void* p) {
    return (unsigned)(unsigned long long)(lds_char_t*)p;
}

__device__ __forceinline__ void wait_tensorcnt0() {
#if __has_builtin(__builtin_amdgcn_s_wait_tensorcnt)
    __builtin_amdgcn_s_wait_tensorcnt(0);
#else
    asm volatile("s_wait_tensorcnt 0x0" ::: "memory");
#endif
}
__device__ __forceinline__ void wait_tensorcnt1() {
#if __has_builtin(__builtin_amdgcn_s_wait_tensorcnt)
    __builtin_amdgcn_s_wait_tensorcnt(1);
#else
    asm volatile("s_wait_tensorcnt 0x1" ::: "memory");
#endif
}

#ifdef HAS_TDM
// Issue one TDM load of a 16-row x 64-col f32 tile (row-major, stride `stride`
// floats) starting at global byte address gaddr into LDS byte offset lds_addr.
// LDS rows padded to 66 dwords (pad_interval=64dw, pad_amount=2dw).
// dim0_rem / rows_rem are the tensor extents measured from the tile start, so
// OOB rows/cols are zero-filled by the TDM.
__device__ __forceinline__ void tdm_issue(unsigned long long gaddr, unsigned lds_addr,
                                          int dim0_rem, int rows_rem, int stride) {
    u32x4 g0;
    g0[0] = 1u;                                   // count=1, user mode
    g0[1] = lds_addr;                             // lds_addr (bytes)
    g0[2] = (unsigned)gaddr;                      // global_addr[31:0]
    g0[3] = (unsigned)(gaddr >> 32) | (2u << 30); // global_addr[56:32] | type=2
    i32x8 g1;
    // data_size=4B (2<<16) | pad_enable (1<<20) | pad_interval=64dw (5<<22)
    // | pad_amount=2dw (1<<25)
    g1[0] = (int)((2u << 16) | (1u << 20) | (5u << 22) | (1u << 25));
    g1[1] = (int)(((unsigned)dim0_rem & 0xFFFFu) << 16);            // tensor_dim0[15:0]
    g1[2] = (int)((((unsigned)dim0_rem >> 16) & 0xFFFFu)            // tensor_dim0[31:16]
                  | (((unsigned)rows_rem & 0xFFFFu) << 16));        // tensor_dim1[15:0]
    g1[3] = (int)((((unsigned)rows_rem >> 16) & 0xFFFFu)            // tensor_dim1[31:16]
                  | (64u << 16));                                   // tile_dim0 = 64
    g1[4] = 16;                                   // tile_dim1 = 16, tile_dim2 = 0
    g1[5] = stride;                               // tensor_dim0_stride[31:0]
    g1[6] = 0;                                    // stride hi / dim1_stride lo
    g1[7] = 0;
    i32x4 gz; gz[0] = 0; gz[1] = 0; gz[2] = 0; gz[3] = 0;
    i32x8 gz8;
    gz8[0] = 0; gz8[1] = 0; gz8[2] = 0; gz8[3] = 0;
    gz8[4] = 0; gz8[5] = 0; gz8[6] = 0; gz8[7] = 0;
    __builtin_amdgcn_tensor_load_to_lds(g0, g1, gz, gz, gz8, 0);
}
#endif

// ---------------------------------------------------------------------------
// GEMV inner loop: one wave accumulates 16 rows of y = W x into a 16x16 f32
// WMMA accumulator (every column of D holds the same 16 dot products).
// A-frag (ISA f32 16x4): lane L<16 : (W[r,kb+0], W[r,kb+1])
//                        lane L>=16: (W[r,kb+2], W[r,kb+3])
// B-frag: same K split, value broadcast across the 16 columns.
// ---------------------------------------------------------------------------
template<int COLS>
__device__ __forceinline__ v8f gemv_tiles(const float* __restrict__ W,
                                          const float* __restrict__ xl,
                                          float* tbuf, int row0, int rows,
                                          int lane, v8f acc) {
    const int m    = lane & 15;
    const int koff = (lane >> 4) << 1;     // 0 or 2
#ifdef HAS_TDM
    constexpr int NT      = COLS / 64;     // k-tiles per matrix
    constexpr int TSTRIDE = 16 * 66;       // floats per LDS tile buffer (padded)
    const int rows_rem = __builtin_amdgcn_readfirstlane(rows - row0);
    const unsigned lds0 =
        (unsigned)__builtin_amdgcn_readfirstlane((int)lds_byte_off(tbuf));
    const unsigned lds1 = lds0 + TSTRIDE * 4;
    const unsigned long long gbase =
        (unsigned long long)(size_t)W +
        (unsigned long long)(unsigned)__builtin_amdgcn_readfirstlane(row0) *
            (unsigned long long)(COLS * 4);

    tdm_issue(gbase, lds0, COLS, rows_rem, COLS);          // tile 0
    for (int t = 0; t < NT; ++t) {
        if (t + 1 < NT) {
            tdm_issue(gbase + (unsigned long long)(t + 1) * 256ull,
                      ((t + 1) & 1) ? lds1 : lds0,
                      COLS - (t + 1) * 64, rows_rem, COLS);
            wait_tensorcnt1();                             // tile t ready
        } else {
            wait_tensorcnt0();
        }
        asm volatile("" ::: "memory");                     // order LDS reads
        const float* buf = tbuf + (t & 1) * TSTRIDE;
        const float2* ap = (const float2*)(buf + m * 66 + koff);  // 8B aligned
        const float2* bp = (const float2*)(xl + t * 64 + koff);
#pragma unroll
        for (int k = 0; k < 64; k += 4) {
            float2 av = ap[k >> 1];
            float2 bv = bp[k >> 1];
            v2f a; a[0] = av.x; a[1] = av.y;
            v2f b; b[0] = bv.x; b[1] = bv.y;
            acc = __builtin_amdgcn_wmma_f32_16x16x4_f32(false, a, false, b,
                                                        (short)0, acc, false, false);
        }
    }
    return acc;
#else
    (void)tbuf; (void)rows;
    const float2* wp = (const float2*)(W + (size_t)(row0 + m) * (size_t)COLS);
    const float2* xp = (const float2*)xl;
    const int idx = koff >> 1;
#pragma unroll 4
    for (int kb2 = 0; kb2 < (COLS >> 1); kb2 += 2) {
        float2 av = wp[kb2 + idx];
        float2 bv = xp[kb2 + idx];
        v2f a; a[0] = av.x; a[1] = av.y;
        v2f b; b[0] = bv.x; b[1] = bv.y;
        acc = __builtin_amdgcn_wmma_f32_16x16x4_f32(false, a, false, b,
                                                    (short)0, acc, false, false);
    }
    return acc;
#endif
}

// y[r] = W[r,:].x (+ W2[r,:].x2) + bias[r] (+ bias2[r]); optional ReLU.
// Block = 128 threads = 4 waves, 16 rows per wave.
template<int COLS, int COLS2, int RELU>
__global__ __launch_bounds__(128)
void gemv_wmma(const float* __restrict__ W,  const float* __restrict__ x,
               const float* __restrict__ bias,
               const float* __restrict__ W2, const float* __restrict__ x2,
               const float* __restrict__ bias2,
               float* __restrict__ y, int rows) {
    __shared__ float xl[COLS];
    __shared__ float x2l[COLS2 > 0 ? COLS2 : 1];
    __shared__ float tiles[4][2 * 16 * 66];

    const int tid = threadIdx.x;
    for (int i = tid; i < COLS; i += 128) xl[i] = x[i];
    if (COLS2 > 0) for (int i = tid; i < COLS2; i += 128) x2l[i] = x2[i];
    __syncthreads();

    const int lane = tid & 31;
    const int wave = tid >> 5;
    const int row0 = blockIdx.x * 64 + wave * 16;
    int row0L = row0;                              // clamped for loads
    if (row0L > rows - 16) row0L = rows - 16;      // all callers have rows>=16
    float* tbuf = tiles[wave];

    v8f acc = {};
    acc = gemv_tiles<COLS>(W, xl, tbuf, row0L, rows, lane, acc);
    if (COLS2 > 0)
        acc = gemv_tiles<(COLS2 > 0 ? COLS2 : 64)>(W2, x2l, tbuf, row0L, rows,
                                                   lane, acc);

    // D layout: VGPR v, lane 0 -> (M=v, N=0); lane 16 -> (M=v+8, N=0)
    if ((lane & 15) == 0) {
        const int base = row0L + ((lane >> 4) << 3);
#pragma unroll
        for (int v = 0; v < 8; ++v) {
            const int rw = base + v;
            if (rw >= row0L && rw < rows) {        // rw in [row0L, rows)
                float val = acc[v] + bias[rw];
                if (COLS2 > 0) val += bias2[rw];
                if (RELU) val = fmaxf(val, 0.0f);
                y[rw] = val;
            }
        }
    }
}

// ---------------------------------------------------------------------------
// Pointwise / small kernels
// ---------------------------------------------------------------------------
__global__ void prep_xa(const int* __restrict__ input,
                        const float* __restrict__ emb,
                        const float* __restrict__ h,
                        float* __restrict__ xa) {
    const int t = blockIdx.x * blockDim.x + threadIdx.x;   // 0..2047
    if (t < HID)            xa[t] = emb[(size_t)input[0] * HID + t];
    else if (t < 2 * HID)   xa[t] = h[t - HID];
}

__global__ void softmax512(const float* __restrict__ z,
                           float* __restrict__ w_ws, float* __restrict__ w_out) {
    __shared__ float red[SEQ];
    const int t = threadIdx.x;
    const float v = z[t];
    red[t] = v; __syncthreads();
    for (int s = SEQ / 2; s > 0; s >>= 1) {
        if (t < s) red[t] = fmaxf(red[t], red[t + s]);
        __syncthreads();
    }
    const float m = red[0]; __syncthreads();
    const float e = expf(v - m);
    red[t] = e; __syncthreads();
    for (int s = SEQ / 2; s > 0; s >>= 1) {
        if (t < s) red[t] += red[t + s];
        __syncthreads();
    }
    const float p = e / red[0];
    w_ws[t] = p;
    w_out[t] = p;
}

// xc[0..H) = embedded ; xc[H..2H) = attn_weights @ encoder_outputs
__global__ void attn_apply(const float* __restrict__ w,
                           const float* __restrict__ enc,
                           const float* __restrict__ emb,
                           const int* __restrict__ input,
                           float* __restrict__ xc) {
    __shared__ float wl[SEQ];
    const int t = threadIdx.x;                 // 0..1023
    if (t < SEQ) wl[t] = w[t];
    __syncthreads();
    float acc = 0.0f;
#pragma unroll 4
    for (int l = 0; l < SEQ; ++l)
        acc = fmaf(wl[l], enc[(size_t)l * HID + t], acc);
    xc[HID + t] = acc;
    xc[t] = emb[(size_t)input[0] * HID + t];
}

__device__ __forceinline__ float sigmoidf(float x) {
    return 1.0f / (1.0f + expf(-x));
}

__global__ void lstm_cell(const float* __restrict__ gates,
                          const float* __restrict__ c0,
                          float* __restrict__ h1_ws, float* __restrict__ c1_ws,
                          float* __restrict__ h1_out, float* __restrict__ c1_out) {
    const int t = threadIdx.x;                 // 0..1023
    const float ig = sigmoidf(gates[t]);
    const float fg = sigmoidf(gates[HID + t]);
    const float gg = tanhf(gates[2 * HID + t]);
    const float og = sigmoidf(gates[3 * HID + t]);
    const float c1 = fg * c0[t] + ig * gg;
    const float h1 = og * tanhf(c1);
    h1_ws[t] = h1; c1_ws[t] = c1;
    h1_out[t] = h1; c1_out[t] = c1;
}

__global__ void logsoftmax(const float* __restrict__ z,
                           float* __restrict__ out, int n) {
    __shared__ float red[1024];
    const int t = threadIdx.x;
    float m = -INFINITY;
    for (int i = t; i < n; i += 1024) m = fmaxf(m, z[i]);
    red[t] = m; __syncthreads();
    for (int s = 512; s > 0; s >>= 1) {
        if (t < s) red[t] = fmaxf(red[t], red[t + s]);
        __syncthreads();
    }
    m = red[0]; __syncthreads();
    float sum = 0.0f;
    for (int i = t; i < n; i += 1024) sum += expf(z[i] - m);
    red[t] = sum; __syncthreads();
    for (int s = 512; s > 0; s >>= 1) {
        if (t < s) red[t] += red[t + s];
        __syncthreads();
    }
    const float lse = m + logf(red[0]);
    for (int i = t; i < n; i += 1024) out[i] = z[i] - lse;
}

// ---------------------------------------------------------------------------
// launch
// ---------------------------------------------------------------------------
extern "C" void kernel_launch(void* const* d_in, const int* in_sizes, int n_in,
                              void* d_out, int out_size, void* d_ws, size_t ws_size,
                              hipStream_t stream) {
    (void)in_sizes; (void)n_in; (void)out_size; (void)ws_size;

    const int*   input  = (const int*)  d_in[0];
    const float* h      = (const float*)d_in[1];
    const float* c0     = (const float*)d_in[2];
    const float* enc    = (const float*)d_in[3];
    const float* emb    = (const float*)d_in[4];
    const float* W_attn = (const float*)d_in[5];
    const float* b_attn = (const float*)d_in[6];
    const float* W_comb = (const float*)d_in[7];
    const float* b_comb = (const float*)d_in[8];
    const float* W_ih   = (const float*)d_in[9];
    const float* W_hh   = (const float*)d_in[10];
    const float* b_ih   = (const float*)d_in[11];
    const float* b_hh   = (const float*)d_in[12];
    const float* W_out  = (const float*)d_in[13];
    const float* b_out  = (const float*)d_in[14];

    float* out = (float*)d_out;
    float* ws  = (float*)d_ws;

    // workspace layout (floats)
    float* xa      = ws;                 // 2048
    float* alog    = xa + 2048;          // 512
    float* aw      = alog + 512;         // 512
    float* xc      = aw + 512;           // 2048
    float* xbuf    = xc + 2048;          // 1024
    float* gates   = xbuf + 1024;        // 4096
    float* h1_ws   = gates + 4096;       // 1024
    float* c1_ws   = h1_ws + 1024;       // 1024
    float* logits  = c1_ws + 1024;       // 50257

    // output layout: logp[V], h1[H], c1[H], attn_w[L]
    float* logp_o = out;
    float* h1_o   = out + VOCAB;
    float* c1_o   = out + VOCAB + HID;
    float* aw_o   = out + VOCAB + 2 * HID;

    const float* nf = nullptr;

    // 1) xa = [embed ; h0]
    prep_xa<<<2, 1024, 0, stream>>>(input, emb, h, xa);
    // 2) attn logits = W_attn @ xa + b_attn              (512 x 2048)
    gemv_wmma<2 * HID, 0, 0><<<8, 128, 0, stream>>>(W_attn, xa, b_attn,
                                                    nf, nf, nf, alog, SEQ);
    // 3) softmax -> attn weights (ws + output)
    softmax512<<<1, SEQ, 0, stream>>>(alog, aw, aw_o);
    // 4) xc = [embed ; attn_w @ encoder_outputs]
    attn_apply<<<1, 1024, 0, stream>>>(aw, enc, emb, input, xc);
    // 5) x = relu(W_comb @ xc + b_comb)                  (1024 x 2048)
    gemv_wmma<2 * HID, 0, 1><<<16, 128, 0, stream>>>(W_comb, xc, b_comb,
                                                     nf, nf, nf, xbuf, HID);
    // 6) gates = W_ih @ x + b_ih + W_hh @ h0 + b_hh      (4096 x 1024, fused)
    gemv_wmma<HID, HID, 0><<<64, 128, 0, stream>>>(W_ih, xbuf, b_ih,
                                                   W_hh, h, b_hh,
                                                   gates, 4 * HID);
    // 7) LSTM cell -> h1, c1
    lstm_cell<<<1, 1024, 0, stream>>>(gates, c0, h1_ws, c1_ws, h1_o, c1_o);
    // 8) logits = W_out @ h1 + b_out                     (50257 x 1024)
    gemv_wmma<HID, 0, 0><<<(VOCAB + 63) / 64, 128, 0, stream>>>(
        W_out, h1_ws, b_out, nf, nf, nf, logits, VOCAB);
    // 9) log-softmax over vocab
    logsoftmax<<<1, 1024, 0, stream>>>(logits, logp_o, VOCAB);
}